// EVETop2MLP_4982162063462
// MI455X (gfx1250) — compile-verified
//
#include <hip/hip_runtime.h>
#include <math.h>

typedef __attribute__((ext_vector_type(2))) float v2f;
typedef __attribute__((ext_vector_type(8))) float v8f;

#define HIDDEN 4096
#define FFN    14336
#define RNK    16
#define TOKENS 4096
#define SCALING 2.0f
#define KS 4            // K-splits for the A-projection (grid.y)
#define FS 4            // ffn-splits for the fused middle (grid.y)
#define HT (TOKENS*16)  // 65536 elements of each [TOKENS,16] tensor

// D = A(16x4 f32) * B(4x16 f32) + C(16x16 f32)
__device__ __forceinline__ v8f wmma4(v2f a, v2f b, v8f c) {
    return __builtin_amdgcn_wmma_f32_16x16x4_f32(
        /*neg_a=*/false, a, /*neg_b=*/false, b,
        /*c_mod=*/(short)0, c, /*reuse_a=*/false, /*reuse_b=*/false);
}

// ---------------------------------------------------------------------------
// Kernel 1: partial h1 = x @ w1_A.T ; partial h3 = x @ w3_A.T
// grid = (T/16, KS), block = 256 (8 waves). Each wave covers 128 K values,
// software-pipelined (distance 2) to keep ~9 b64 loads in flight.
// ---------------------------------------------------------------------------
__global__ void lora_a_proj(const float* __restrict__ x,
                            const float* __restrict__ w1A,
                            const float* __restrict__ w3A,
                            float* __restrict__ p1, float* __restrict__ p3) {
    __shared__ float s1[256];
    __shared__ float s3[256];
    const int tid  = threadIdx.x;
    const int wave = tid >> 5;
    const int lane = tid & 31;
    const int m    = lane & 15;
    const int koff = (lane >> 4) * 2;   // lanes 16..31 carry K+2,K+3
    const int mhi  = (lane >> 4) * 8;   // D-layout: hi half holds M+8
    const int tile = blockIdx.x;

    s1[tid] = 0.0f;
    s3[tid] = 0.0f;
    __syncthreads();

    const int t0 = tile * 16;
    const float* xrow  = x   + (size_t)(t0 + m) * HIDDEN + koff;  // A[m][k]
    const float* a1row = w1A + (size_t)m * HIDDEN + koff;         // B[k][n] = w1A[n][k]
    const float* a3row = w3A + (size_t)m * HIDDEN + koff;

    const int kbeg = (blockIdx.y * 8 + wave) * (HIDDEN / (KS * 8)); // 128 K per wave
    const int kend = kbeg + HIDDEN / (KS * 8);

    v8f acc1 = {};
    v8f acc3 = {};
    // distance-2 software pipeline
    v2f a_c  = *(const v2f*)(xrow  + kbeg);
    v2f b1_c = *(const v2f*)(a1row + kbeg);
    v2f b3_c = *(const v2f*)(a3row + kbeg);
    v2f a_n  = *(const v2f*)(xrow  + kbeg + 4);
    v2f b1_n = *(const v2f*)(a1row + kbeg + 4);
    v2f b3_n = *(const v2f*)(a3row + kbeg + 4);
    for (int k0 = kbeg; k0 + 8 < kend; k0 += 4) {
        v2f a_p  = *(const v2f*)(xrow  + k0 + 8);
        v2f b1_p = *(const v2f*)(a1row + k0 + 8);
        v2f b3_p = *(const v2f*)(a3row + k0 + 8);
        acc1 = wmma4(a_c, b1_c, acc1);
        acc3 = wmma4(a_c, b3_c, acc3);
        a_c = a_n;  b1_c = b1_n;  b3_c = b3_n;
        a_n = a_p;  b1_n = b1_p;  b3_n = b3_p;
    }
    acc1 = wmma4(a_c, b1_c, acc1);
    acc3 = wmma4(a_c, b3_c, acc3);
    acc1 = wmma4(a_n, b1_n, acc1);
    acc3 = wmma4(a_n, b3_n, acc3);

#pragma unroll
    for (int j = 0; j < 8; ++j) {
        const int mm = j + mhi;
        atomicAdd(&s1[mm * 16 + m], acc1[j]);
        atomicAdd(&s3[mm * 16 + m], acc3[j]);
    }
    __syncthreads();
    const size_t o = (size_t)blockIdx.y * HT + (size_t)tile * 256 + tid;
    p1[o] = s1[tid];
    p3[o] = s3[tid];
}

// Deterministic reduce of the KS partials + bias -> h1, h3
__global__ void reduce_a(const float* __restrict__ p1, const float* __restrict__ p3,
                         const float* __restrict__ w1Ab, const float* __restrict__ w3Ab,
                         float* __restrict__ h1g, float* __restrict__ h3g) {
    const int i = blockIdx.x * 256 + threadIdx.x;   // 0..HT-1
    float a = w1Ab[i & 15];
    float b = w3Ab[i & 15];
#pragma unroll
    for (int k = 0; k < KS; ++k) {
        a += p1[(size_t)k * HT + i];
        b += p3[(size_t)k * HT + i];
    }
    h1g[i] = a;
    h3g[i] = b;
}

// ---------------------------------------------------------------------------
// Kernel 2: fused  g=(h1@w1_B.T+b)*2 ; u=(h3@w3_B.T+b)*2 ; h=silu(g)*u ;
//           partial h2 = h @ w2_A.T   (h never touches HBM)
// grid = (T/16, FS), block = 128 (4 waves); ffn chunks strided by FS*4.
// ---------------------------------------------------------------------------
#define LP 18  // LDS row pitch (even for b64 alignment, bank-conflict-free)

__global__ void lora_mid_fused(const float* __restrict__ h1g, const float* __restrict__ h3g,
                               const float* __restrict__ w1B, const float* __restrict__ w1Bb,
                               const float* __restrict__ w3B, const float* __restrict__ w3Bb,
                               const float* __restrict__ w2A,
                               float* __restrict__ ph2) {
    __shared__ float s1[16 * LP];
    __shared__ float s3[16 * LP];
    __shared__ float hbuf[4][16 * LP];
    __shared__ float s2[256];
    const int tid  = threadIdx.x;       // 128
    const int wave = tid >> 5;
    const int lane = tid & 31;
    const int n    = lane & 15;
    const int koff = (lane >> 4) * 2;
    const int mhi  = (lane >> 4) * 8;
    const int tile = blockIdx.x;

    for (int i = tid; i < 256; i += 128) {
        const int mm = i >> 4, rr = i & 15;
        s1[mm * LP + rr] = h1g[(size_t)tile * 256 + i];
        s3[mm * LP + rr] = h3g[(size_t)tile * 256 + i];
        s2[i] = 0.0f;
    }
    __syncthreads();

    // Hoist the loop-invariant A-fragments (h1/h3 tiles) into registers so the
    // asm memory clobber below cannot force per-iteration LDS reloads.
    v2f a1f[4], a3f[4];
#pragma unroll
    for (int i = 0; i < 4; ++i) {
        a1f[i] = *(const v2f*)(&s1[n * LP + i * 4 + koff]);
        a3f[i] = *(const v2f*)(&s3[n * LP + i * 4 + koff]);
    }

    float* hb = hbuf[wave];
    v8f h2acc = {};

    for (int fc = blockIdx.y * 4 + wave; fc < FFN / 16; fc += FS * 4) {
        const int f0 = fc * 16;
        const int f  = f0 + n;
        v8f g = {};
        v8f u = {};
#pragma unroll
        for (int i = 0; i < 4; ++i) {
            const int k0 = i * 4;
            v2f b1 = *(const v2f*)(&w1B[(size_t)f * RNK + k0 + koff]);  // B[k][n]=w1B[f][k]
            v2f b3 = *(const v2f*)(&w3B[(size_t)f * RNK + k0 + koff]);
            g = wmma4(a1f[i], b1, g);
            u = wmma4(a3f[i], b3, u);
        }
        const float gb = w1Bb[f];
        const float ub = w3Bb[f];
#pragma unroll
        for (int j = 0; j < 8; ++j) {
            const float gg = (g[j] + gb) * SCALING;
            const float uu = (u[j] + ub) * SCALING;
            const float hh = (gg / (1.0f + __expf(-gg))) * uu;     // silu(g)*u
            hb[(j + mhi) * LP + n] = hh;                           // D-layout -> LDS tile
        }
        asm volatile("s_wait_dscnt 0" ::: "memory");               // same-wave LDS RAW
#pragma unroll
        for (int k0 = 0; k0 < 16; k0 += 4) {
            v2f a = *(const v2f*)(&hb[n * LP + k0 + koff]);        // A[m=tok][k=f_local]
            v2f b = *(const v2f*)(&w2A[(size_t)n * FFN + f0 + k0 + koff]); // B[k][n]=w2A[r][f0+k]
            h2acc = wmma4(a, b, h2acc);
        }
    }

#pragma unroll
    for (int j = 0; j < 8; ++j)
        atomicAdd(&s2[(j + mhi) * 16 + n], h2acc[j]);
    __syncthreads();
    for (int i = tid; i < 256; i += 128)
        ph2[(size_t)blockIdx.y * HT + (size_t)tile * 256 + i] = s2[i];
}

// Deterministic reduce of the FS partials + bias -> h2
__global__ void reduce_h2(const float* __restrict__ ph2, const float* __restrict__ w2Ab,
                          float* __restrict__ h2g) {
    const int i = blockIdx.x * 256 + threadIdx.x;
    float a = w2Ab[i & 15];
#pragma unroll
    for (int k = 0; k < FS; ++k)
        a += ph2[(size_t)k * HT + i];
    h2g[i] = a;
}

// ---------------------------------------------------------------------------
// Kernel 3: out = (h2 @ w2_B.T + w2_Bb) * 2    ([T, HIDDEN], write-bound)
// grid = (T/16, 32), block = 256 (8 waves): one 16x16 out tile per wave.
// ---------------------------------------------------------------------------
__global__ void lora_b_out(const float* __restrict__ h2g,
                           const float* __restrict__ w2B, const float* __restrict__ w2Bb,
                           float* __restrict__ out) {
    const int tid  = threadIdx.x;
    const int wave = tid >> 5;
    const int lane = tid & 31;
    const int n    = lane & 15;
    const int koff = (lane >> 4) * 2;
    const int mhi  = (lane >> 4) * 8;
    const int tile  = blockIdx.x;                 // token tile
    const int ntile = blockIdx.y * 8 + wave;      // output tile along HIDDEN
    const int t0 = tile * 16;
    const int n0 = ntile * 16;

    const float* arow = h2g + (size_t)tile * 256 + n * 16;   // A[m=n][k=r]
    const float* brow = w2B + (size_t)(n0 + n) * RNK;        // B[k][n] = w2B[n0+n][k]
    v8f acc = {};
#pragma unroll
    for (int k0 = 0; k0 < RNK; k0 += 4) {
        v2f a = *(const v2f*)(arow + k0 + koff);
        v2f b = *(const v2f*)(brow + k0 + koff);
        acc = wmma4(a, b, acc);
    }
    const float bias = w2Bb[n0 + n];
#pragma unroll
    for (int j = 0; j < 8; ++j)
        out[(size_t)(t0 + j + mhi) * HIDDEN + n0 + n] = (acc[j] + bias) * SCALING;
}

// ---------------------------------------------------------------------------
extern "C" void kernel_launch(void* const* d_in, const int* in_sizes, int n_in,
                              void* d_out, int out_size, void* d_ws, size_t ws_size,
                              hipStream_t stream) {
    const float* x     = (const float*)d_in[0];
    const float* w1A   = (const float*)d_in[1];
    const float* w1Ab  = (const float*)d_in[2];
    const float* w1B   = (const float*)d_in[3];
    const float* w1Bb  = (const float*)d_in[4];
    const float* w3A   = (const float*)d_in[5];
    const float* w3Ab  = (const float*)d_in[6];
    const float* w3B   = (const float*)d_in[7];
    const float* w3Bb  = (const float*)d_in[8];
    const float* w2A   = (const float*)d_in[9];
    const float* w2Ab  = (const float*)d_in[10];
    const float* w2B   = (const float*)d_in[11];
    const float* w2Bb  = (const float*)d_in[12];
    float* out = (float*)d_out;

    float* wsf = (float*)d_ws;
    float* p1  = wsf;                      // [KS][HT]
    float* p3  = p1 + (size_t)KS * HT;     // [KS][HT]
    float* h1g = p3 + (size_t)KS * HT;     // [HT]
    float* h3g = h1g + HT;                 // [HT]
    float* ph2 = h3g + HT;                 // [FS][HT]
    float* h2g = ph2 + (size_t)FS * HT;    // [HT]

    const int ttiles = TOKENS / 16;        // 256

    lora_a_proj<<<dim3(ttiles, KS), 256, 0, stream>>>(x, w1A, w3A, p1, p3);
    reduce_a<<<HT / 256, 256, 0, stream>>>(p1, p3, w1Ab, w3Ab, h1g, h3g);
    lora_mid_fused<<<dim3(ttiles, FS), 128, 0, stream>>>(h1g, h3g, w1B, w1Bb,
                                                         w3B, w3Bb, w2A, ph2);
    reduce_h2<<<HT / 256, 256, 0, stream>>>(ph2, w2Ab, h2g);
    lora_b_out<<<dim3(ttiles, 32), 256, 0, stream>>>(h2g, w2B, w2Bb, out);
}